// NClosestThresholdVerificationHead_11166914970468
// MI455X (gfx1250) — compile-verified
//
#include <hip/hip_runtime.h>
#include <hip/hip_bf16.h>

typedef __attribute__((ext_vector_type(2))) float v2f;
typedef __attribute__((ext_vector_type(8))) float v8f;

#define DIM 128

// ---------------------------------------------------------------------------
// Kernel 1: squared L2 norms of every query row and reference row -> d_ws
// ---------------------------------------------------------------------------
__global__ void nclosest_norms_kernel(const float* __restrict__ Qm,
                                      const float* __restrict__ Rm,
                                      float* __restrict__ qsq,
                                      float* __restrict__ rsq,
                                      int Q, int R) {
    int gid = blockIdx.x * blockDim.x + threadIdx.x;
    const float* row;
    float* dst;
    if (gid < Q) {
        row = Qm + (size_t)gid * DIM;
        dst = qsq + gid;
    } else if (gid < Q + R) {
        int r = gid - Q;
        row = Rm + (size_t)r * DIM;
        dst = rsq + r;
    } else {
        return;
    }
    const float4* p = (const float4*)row;
    float s = 0.0f;
#pragma unroll
    for (int i = 0; i < DIM / 4; ++i) {
        float4 v = p[i];
        s += v.x * v.x + v.y * v.y + v.z * v.z + v.w * v.w;
    }
    *dst = s;
}

// ---------------------------------------------------------------------------
// Kernel 2: WMMA Gram-matrix threshold count.
//   block = 256 threads (8 wave32), one block per 16-query strip.
//   Each wave: A strip (16x128 f32) resident in VGPRs, sweeps R/8 refs in
//   PAIRS of 16-wide tiles with two independent accumulator chains so the
//   v_wmma_f32_16x16x4_f32 issue stream never stalls on its own D->C RAW.
//   Epilogue: sq = q^2 + r^2 - 2*dot <= t^2 -> per-lane counters,
//   LDS atomic reduce across waves, out = min(count,100)/100.
// ---------------------------------------------------------------------------
__launch_bounds__(256)
__global__ void nclosest_gram_count_kernel(const float* __restrict__ Qm,
                                           const float* __restrict__ Rm,
                                           const float* __restrict__ qsq,
                                           const float* __restrict__ rsq,
                                           const float* __restrict__ thr,
                                           float* __restrict__ out,
                                           int R) {
    const int lane = threadIdx.x & 31;   // wave32
    const int wave = threadIdx.x >> 5;   // 0..7
    const int x    = lane & 15;          // row within tile (A) / col (B,C,D)
    const int h    = lane >> 4;          // lane-half selects K pair / M half
    const int qBase = blockIdx.x * 16;

    // Preload A fragments: lane holds Q[qBase+x][4*ks + 2*h + {0,1}] per step.
    v2f a[DIM / 4];
    {
        const float* qrow = Qm + (size_t)(qBase + x) * DIM + 2 * h;
#pragma unroll
        for (int ks = 0; ks < DIM / 4; ++ks)
            a[ks] = *(const v2f*)(qrow + 4 * ks);
    }

    const float t  = thr[0];
    const float t2 = t * t;

    // q^2 for this lane's 8 output rows (m = 8*h + v), contiguous.
    float qs[8];
#pragma unroll
    for (int v = 0; v < 8; ++v) qs[v] = qsq[qBase + 8 * h + v];

    int cnt[8];
#pragma unroll
    for (int v = 0; v < 8; ++v) cnt[v] = 0;

    const int tilesPerWave = R / (16 * 8);     // 128 for R=16384
    const int tile0 = wave * tilesPerWave;

    for (int ti = 0; ti < tilesPerWave; ti += 2) {
        const int rBase0 = (tile0 + ti) * 16;
        const int rBase1 = rBase0 + 16;
        const float* rrow0 = Rm + (size_t)(rBase0 + x) * DIM + 2 * h;
        const float* rrow1 = Rm + (size_t)(rBase1 + x) * DIM + 2 * h;

        v8f acc0 = {0.f, 0.f, 0.f, 0.f, 0.f, 0.f, 0.f, 0.f};
        v8f acc1 = {0.f, 0.f, 0.f, 0.f, 0.f, 0.f, 0.f, 0.f};
#pragma unroll
        for (int ks = 0; ks < DIM / 4; ++ks) {
            v2f b0 = *(const v2f*)(rrow0 + 4 * ks);
            v2f b1 = *(const v2f*)(rrow1 + 4 * ks);
            // Two independent D = A x B + C chains sharing the A fragment.
            acc0 = __builtin_amdgcn_wmma_f32_16x16x4_f32(
                false, a[ks], false, b0, (short)0, acc0, false, false);
            acc1 = __builtin_amdgcn_wmma_f32_16x16x4_f32(
                false, a[ks], false, b1, (short)0, acc1, false, false);
        }

        const float rn0 = rsq[rBase0 + x];   // this lane's column n = x
        const float rn1 = rsq[rBase1 + x];
#pragma unroll
        for (int v = 0; v < 8; ++v) {
            float sq0 = qs[v] + rn0 - 2.0f * acc0[v];
            float sq1 = qs[v] + rn1 - 2.0f * acc1[v];
            cnt[v] += (sq0 <= t2) ? 1 : 0;
            cnt[v] += (sq1 <= t2) ? 1 : 0;
        }
    }

    // Reduce across lanes and waves.
    __shared__ int scnt[16];
    if (threadIdx.x < 16) scnt[threadIdx.x] = 0;
    __syncthreads();
#pragma unroll
    for (int v = 0; v < 8; ++v)
        atomicAdd(&scnt[8 * h + v], cnt[v]);
    __syncthreads();

    if (threadIdx.x < 16) {
        int c = scnt[threadIdx.x];
        if (c > 100) c = 100;
        out[qBase + threadIdx.x] = (float)c * 0.01f;
    }
}

// ---------------------------------------------------------------------------
extern "C" void kernel_launch(void* const* d_in, const int* in_sizes, int n_in,
                              void* d_out, int out_size, void* d_ws, size_t ws_size,
                              hipStream_t stream) {
    const float* Qm  = (const float*)d_in[0];
    const float* Rm  = (const float*)d_in[1];
    const float* thr = (const float*)d_in[2];
    float* out = (float*)d_out;

    const int Q = in_sizes[0] / DIM;   // 4096
    const int R = in_sizes[1] / DIM;   // 16384

    float* qsq = (float*)d_ws;
    float* rsq = qsq + Q;

    {
        int total = Q + R;
        int blocks = (total + 255) / 256;
        nclosest_norms_kernel<<<blocks, 256, 0, stream>>>(Qm, Rm, qsq, rsq, Q, R);
    }
    {
        int blocks = Q / 16;   // one 16-query strip per block
        nclosest_gram_count_kernel<<<blocks, 256, 0, stream>>>(Qm, Rm, qsq, rsq,
                                                               thr, out, R);
    }
}